// loss_56684978372843
// MI455X (gfx1250) — compile-verified
//
#include <hip/hip_runtime.h>

// Problem constants (fixed by the reference harness).
#define BB 8
#define AA 65536
#define CC 80
#define MM 32
#define NB1 (BB * AA / 256)   // prep blocks: 2048 (256 per image)
#define NB2 (BB * AA / 64)    // focal blocks: 8192 (1024 per image, 64 anchors each)

typedef float v2f __attribute__((ext_vector_type(2)));
typedef float v4f __attribute__((ext_vector_type(4)));
typedef float v8f __attribute__((ext_vector_type(8)));

// ---------------------------------------------------------------------------
// Wave32 sum reduction via V_WMMA_F32_16X16X4_F32.
// A (16x4) holds each lane's scalar in VGPR0 (K=0 for lanes 0-15, K=2 for
// lanes 16-31), VGPR1 = 0. B = all ones (layout-independent). Then
// D[m][n] = acc_m + acc_{m+16} for every column n. Summing the 8 D VGPRs in
// each lane gives rows 0-7 (lanes<16) / rows 8-15 (lanes>=16) column sums;
// one xor-16 shuffle adds the two halves -> total in every lane.
// EXEC must be all ones at the call site (all 256 threads alive).
// ---------------------------------------------------------------------------
__device__ __forceinline__ float wave_sum32(float x) {
#if __has_builtin(__builtin_amdgcn_wmma_f32_16x16x4_f32)
  v2f a;  a.x = x;   a.y = 0.0f;
  v2f bo; bo.x = 1.0f; bo.y = 1.0f;
  v8f c = {};
  v8f d = __builtin_amdgcn_wmma_f32_16x16x4_f32(false, a, false, bo,
                                                (short)0, c, false, false);
  float s = d[0] + d[1] + d[2] + d[3] + d[4] + d[5] + d[6] + d[7];
  s += __shfl_xor(s, 16, 32);
  return s;
#else
  float s = x;
  for (int off = 16; off > 0; off >>= 1) s += __shfl_xor(s, off, 32);
  return s;
#endif
}

__device__ __forceinline__ float smoothl1(float d) {
  float ad = fabsf(d);
  return (ad <= 1.0f) ? 0.5f * ad * ad : ad - 0.5f;
}

// ---------------------------------------------------------------------------
// Kernel 1: per-anchor IoU/argmax vs 32 boxes, pack info, smooth-L1 for pos.
// 256 threads = 256 anchors per block. 2048 blocks.
// info bits: [0]=pos, [1]=neg, [15:8]=target class
// ---------------------------------------------------------------------------
__global__ void __launch_bounds__(256)
prep_kernel(const float* __restrict__ anchors, const float* __restrict__ reg,
            const float* __restrict__ labels,
            int* __restrict__ info, int* __restrict__ posPart,
            float* __restrict__ regPart) {
  const int blocksPerImg = AA / 256;
  const int b  = blockIdx.x / blocksPerImg;
  const int a0 = (blockIdx.x % blocksPerImg) * 256;
  const int t  = threadIdx.x;

  __shared__ float bx[MM][4];
  __shared__ int   bcls[MM];
  __shared__ float wsumF[8];
  __shared__ int   wsumI[8];

  if (t < MM) {
    const float* L = labels + ((size_t)b * MM + t) * 6;
    bcls[t]  = (int)L[1];
    bx[t][0] = L[2]; bx[t][1] = L[3]; bx[t][2] = L[4]; bx[t][3] = L[5];
  }
  __syncthreads();

  const int a = a0 + t;
  const v4f an =
      __builtin_nontemporal_load(((const v4f*)(anchors + (size_t)b * AA * 4)) + a);
  const float areaA = (an.z - an.x) * (an.w - an.y);

  float best = -1.0f; int bestId = 0;
  #pragma unroll 8
  for (int j = 0; j < MM; ++j) {
    float lx = fmaxf(an.x, bx[j][0]);
    float ly = fmaxf(an.y, bx[j][1]);
    float rx = fminf(an.z, bx[j][2]);
    float ry = fminf(an.w, bx[j][3]);
    float w  = fmaxf(rx - lx, 0.0f);
    float h  = fmaxf(ry - ly, 0.0f);
    float inter = w * h;
    float areaB = (bx[j][2] - bx[j][0]) * (bx[j][3] - bx[j][1]);
    float iou = inter / (areaA + areaB - inter);
    if (iou > best) { best = iou; bestId = j; }  // strict > keeps first argmax
  }

  const bool pos = best > 0.5f;
  const bool neg = best < 0.4f;
  info[(size_t)b * AA + a] = (pos ? 1 : 0) | (neg ? 2 : 0) | (bcls[bestId] << 8);

  float regLoss = 0.0f;
  if (pos) {
    v4f r =
        __builtin_nontemporal_load(((const v4f*)(reg + (size_t)b * AA * 4)) + a);
    float gx0 = bx[bestId][0], gy0 = bx[bestId][1];
    float gx1 = bx[bestId][2], gy1 = bx[bestId][3];
    float ax = (an.x + an.z) * 0.5f, ay = (an.y + an.w) * 0.5f;
    float aw = an.z - an.x,          ah = an.w - an.y;
    float gx = (gx0 + gx1) * 0.5f,   gy = (gy0 + gy1) * 0.5f;
    float gw = gx1 - gx0,            gh = gy1 - gy0;
    regLoss = smoothl1(r.x - (gx - ax) / aw) +
              smoothl1(r.y - (gy - ay) / ah) +
              smoothl1(r.z - __logf(gw / aw)) +
              smoothl1(r.w - __logf(gh / ah));
  }

  // Block reductions (deterministic).
  const int wave = t >> 5, lane = t & 31;
  unsigned long long bal = __ballot(pos);
  int wcnt = __popcll(bal);
  float wsum = wave_sum32(regLoss);
  if (lane == 0) { wsumF[wave] = wsum; wsumI[wave] = wcnt; }
  __syncthreads();
  if (t == 0) {
    float fr = 0.0f; int ci = 0;
    #pragma unroll
    for (int i = 0; i < 8; ++i) { fr += wsumF[i]; ci += wsumI[i]; }
    regPart[blockIdx.x] = fr;
    posPart[blockIdx.x] = ci;
  }
}

// ---------------------------------------------------------------------------
// Kernel 2: focal loss over cls (B x A x C). Each block: 64 anchors x 80
// classes = 1280 float4s; 256 threads x 5 float4s each. 8192 blocks.
//
// Branch-free single-log focal term:
//   q = isTgt ? p : 1-p,  coef = isTgt ? ALPHA : 1-ALPHA
//   term = coef * (1-q)^2 * (-log q)
// which equals ALPHA*(1-p)^2*(-log p) when tgt=1 and
// (1-ALPHA)*p^2*(-log(1-p)) when tgt=0  -> exactly one v_log_f32 per element.
// ---------------------------------------------------------------------------
__global__ void __launch_bounds__(256)
focal_kernel(const float* __restrict__ cls, const int* __restrict__ info,
             float* __restrict__ clsPart) {
  const int blocksPerImg = AA / 64;
  const int b     = blockIdx.x / blocksPerImg;
  const int aBase = (blockIdx.x % blocksPerImg) * 64;
  const int t     = threadIdx.x;

  __shared__ int   sInfo[64];
  __shared__ float wsumF[8];
  if (t < 64) sInfo[t] = info[(size_t)b * AA + aBase + t];
  __syncthreads();

  const v4f* base =
      (const v4f*)(cls + ((size_t)b * AA + aBase) * CC);  // 320B aligned

  float acc = 0.0f;
  #pragma unroll
  for (int i = 0; i < 5; ++i) {
    const int f    = t + 256 * i;      // float4 index within block: 0..1279
    const int aLoc = f / 20;           // 20 float4s per anchor (C=80)
    const int inf  = sInfo[aLoc];
    v4f v = __builtin_nontemporal_load(base + f);  // single-pass stream: TH=NT
    if ((inf & 3) != 0) {              // pos or neg => included
      const bool posA = (inf & 1) != 0;
      const int  tcls = inf >> 8;
      const int  cB   = (f % 20) * 4;
      float pv[4] = {v.x, v.y, v.z, v.w};
      #pragma unroll
      for (int k = 0; k < 4; ++k) {
        float p = fminf(fmaxf(pv[k], 1.0e-4f), 1.0f - 1.0e-4f);
        const bool isTgt = posA && ((cB + k) == tcls);
        float q    = isTgt ? p : 1.0f - p;
        float coef = isTgt ? 0.25f : 0.75f;
        float r    = 1.0f - q;
        acc += coef * r * r * (-__logf(q));
      }
    }
  }

  const int wave = t >> 5, lane = t & 31;
  float wsum = wave_sum32(acc);
  if (lane == 0) wsumF[wave] = wsum;
  __syncthreads();
  if (t == 0) {
    float s = 0.0f;
    #pragma unroll
    for (int i = 0; i < 8; ++i) s += wsumF[i];
    clsPart[blockIdx.x] = s;
  }
}

// ---------------------------------------------------------------------------
// Kernel 3: final deterministic reduction. 1 block; thread b handles image b.
// ---------------------------------------------------------------------------
__global__ void __launch_bounds__(64)
finalize_kernel(const int* __restrict__ posPart, const float* __restrict__ regPart,
                const float* __restrict__ clsPart, float* __restrict__ out) {
  __shared__ float acc[BB];
  const int t = threadIdx.x;
  if (t < BB) {
    const int pb = NB1 / BB;   // 256 prep blocks per image
    const int fb = NB2 / BB;   // 1024 focal blocks per image
    int   np = 0; float rs = 0.0f;
    for (int i = 0; i < pb; ++i) { np += posPart[t * pb + i]; rs += regPart[t * pb + i]; }
    float cs = 0.0f;
    for (int i = 0; i < fb; ++i) cs += clsPart[t * fb + i];
    float npf = (float)(np > 1 ? np : 1);
    acc[t] = (cs + rs) / npf;
  }
  __syncthreads();
  if (t == 0) {
    float s = 0.0f;
    #pragma unroll
    for (int i = 0; i < BB; ++i) s += acc[i];
    out[0] = s / (float)BB;
  }
}

// ---------------------------------------------------------------------------
extern "C" void kernel_launch(void* const* d_in, const int* in_sizes, int n_in,
                              void* d_out, int out_size, void* d_ws, size_t ws_size,
                              hipStream_t stream) {
  const float* cls     = (const float*)d_in[0];  // B*A*C
  const float* reg     = (const float*)d_in[1];  // B*A*4
  const float* labels  = (const float*)d_in[2];  // B*M*6
  const float* anchors = (const float*)d_in[3];  // B*A*4
  float* out = (float*)d_out;

  char* ws = (char*)d_ws;
  int*   info    = (int*)ws;                                    // B*A ints (2 MB)
  int*   posPart = (int*)(ws + (size_t)BB * AA * 4);            // NB1 ints
  float* regPart = (float*)(ws + (size_t)BB * AA * 4 + NB1 * 4);
  float* clsPart = (float*)(ws + (size_t)BB * AA * 4 + NB1 * 8);

  prep_kernel<<<NB1, 256, 0, stream>>>(anchors, reg, labels, info, posPart, regPart);
  focal_kernel<<<NB2, 256, 0, stream>>>(cls, info, clsPart);
  finalize_kernel<<<1, 64, 0, stream>>>(posPart, regPart, clsPart, out);
}